// YOLOLoss_34608846471441
// MI455X (gfx1250) — compile-verified
//
#include <hip/hip_runtime.h>

// ---------------- Problem constants ----------------
#define S_GRID 7
#define N_IMG  16384
#define D_FEAT 30                       // B*5 + C = 2*5 + 20
#define NCELLS (N_IMG * S_GRID * S_GRID) // 802816
#define LAMBDA_COORD 5.0f
#define LAMBDA_NOOBJ 0.5f

#define NBLOCKS 1024
#define NTHREADS 256

typedef float v2f __attribute__((ext_vector_type(2)));
typedef float v8f __attribute__((ext_vector_type(8)));

// ---------------- Main streaming pass ----------------
// One thread per cell (grid-stride). Accumulates 4 partial losses:
// [0]=coord, [1]=obj, [2]=noobj, [3]=class. Block partials -> ws[blk*4+q].
__global__ __launch_bounds__(NTHREADS)
void yolo_loss_main(const float* __restrict__ pred,
                    const float* __restrict__ tgt,
                    float* __restrict__ ws)
{
    float acc0 = 0.f, acc1 = 0.f, acc2 = 0.f, acc3 = 0.f;

    const int stride = NBLOCKS * NTHREADS;
    for (int c = blockIdx.x * NTHREADS + threadIdx.x; c < NCELLS; c += stride) {
        const v2f* p2 = reinterpret_cast<const v2f*>(pred + (size_t)c * D_FEAT);
        const v2f* t2 = reinterpret_cast<const v2f*>(tgt  + (size_t)c * D_FEAT);

        float p[30];
        float t[30];
        // predictions: all 15 float2 chunks (stream-once -> non-temporal)
        #pragma unroll
        for (int i = 0; i < 15; ++i) {
            v2f v = __builtin_nontemporal_load(&p2[i]);
            p[2 * i] = v[0]; p[2 * i + 1] = v[1];
        }
        // targets: skip chunks 3,4 (floats 6..9 are unused by the loss)
        #pragma unroll
        for (int i = 0; i < 15; ++i) {
            if (i == 3 || i == 4) continue;
            v2f v = __builtin_nontemporal_load(&t2[i]);
            t[2 * i] = v[0]; t[2 * i + 1] = v[1];
        }

        const float tw = t[2], th = t[3], tconf = t[4];
        const float ta = tw * th;

        // IoU for both boxes against the (single) target box
        const float pw0 = p[2], ph0 = p[3];
        const float pw1 = p[7], ph1 = p[8];
        const float in0 = fminf(pw0, tw) * fminf(ph0, th);
        const float in1 = fminf(pw1, tw) * fminf(ph1, th);
        const float u0  = pw0 * ph0 + ta - in0;
        const float u1  = pw1 * ph1 + ta - in1;
        const float iou0 = in0 / (u0 + 1e-6f);
        const float iou1 = in1 / (u1 + 1e-6f);
        const bool best1 = iou1 > iou0;          // argmax: first max wins ties -> box0
        const bool obj   = (tconf == 1.0f);
        const bool noobj = (tconf == 0.0f);

        // confidence SSE over both boxes
        const float dq0 = p[4] - tconf;
        const float dq1 = p[9] - tconf;
        const float sq  = dq0 * dq0 + dq1 * dq1;
        if (obj)   acc1 += sq;
        if (noobj) acc2 += sq;

        if (obj) {
            // coordinate loss on the responsible box
            const int o = best1 ? 5 : 0;
            const float d0 = p[o + 0] - t[0];
            const float d1 = p[o + 1] - t[1];
            const float d2 = p[o + 2] - t[2];
            const float d3 = p[o + 3] - t[3];
            acc0 += d0 * d0 + d1 * d1 + d2 * d2 + d3 * d3;

            // class BCE (only contributes for obj cells)
            float cl = 0.f;
            #pragma unroll
            for (int k = 0; k < 20; ++k) {
                const float pc = p[10 + k];
                const float tc = t[10 + k];
                const float lp = fmaxf(logf(pc),      -100.0f);
                const float lm = fmaxf(log1pf(-pc),   -100.0f);
                cl += -(tc * lp + (1.0f - tc) * lm);
            }
            acc3 += cl;
        }
    }

    // ---- wave32 reduction ----
    #pragma unroll
    for (int off = 16; off > 0; off >>= 1) {
        acc0 += __shfl_xor(acc0, off, 32);
        acc1 += __shfl_xor(acc1, off, 32);
        acc2 += __shfl_xor(acc2, off, 32);
        acc3 += __shfl_xor(acc3, off, 32);
    }

    __shared__ float lds[NTHREADS / 32][4];   // 8 waves per block
    const int lane = threadIdx.x & 31;
    const int wave = threadIdx.x >> 5;
    if (lane == 0) {
        lds[wave][0] = acc0; lds[wave][1] = acc1;
        lds[wave][2] = acc2; lds[wave][3] = acc3;
    }
    __syncthreads();
    if (threadIdx.x < 4) {
        float s = 0.f;
        #pragma unroll
        for (int w = 0; w < NTHREADS / 32; ++w) s += lds[w][threadIdx.x];
        ws[blockIdx.x * 4 + threadIdx.x] = s;
    }
}

// ---------------- Final WMMA reduction ----------------
// Single wave32. Reduces ws[1024][4] with chained V_WMMA_F32_16X16X4_F32:
// D += A(16x4) x Ones(4x16). With B = all-ones the result depends only on the
// documented A layout (lanes 0-15: K=0,1 ; lanes 16-31: K=2,3 ; M = lane%16):
// D[m][n] accumulates sum_k A[m][k], i.e. 64 partial rows per WMMA issue.
__global__ __launch_bounds__(32)
void yolo_loss_final(const float* __restrict__ ws, float* __restrict__ out)
{
    const int lane  = threadIdx.x;        // 0..31, EXEC all ones through WMMAs
    const int m     = lane & 15;
    const int kbase = (lane >> 4) * 2;    // 0 or 2

    v2f bones; bones[0] = 1.0f; bones[1] = 1.0f;

    float tot[4];
    #pragma unroll
    for (int q = 0; q < 4; ++q) {
        v8f c = {};
        #pragma unroll
        for (int tile = 0; tile < NBLOCKS / 64; ++tile) {   // 16 chained WMMAs
            const int row = tile * 64 + m * 4 + kbase;      // A[m][kbase+v] = ws[row+v][q]
            v2f a;
            a[0] = ws[(row + 0) * 4 + q];
            a[1] = ws[(row + 1) * 4 + q];
            c = __builtin_amdgcn_wmma_f32_16x16x4_f32(
                    /*neg_a=*/false, a, /*neg_b=*/false, bones,
                    /*c_mod=*/(short)0, c, /*reuse_a=*/false, /*reuse_b=*/false);
        }
        // lanes 0-15 hold D[m=0..7][n], lanes 16-31 hold D[m=8..15][n]
        float s = c[0] + c[1] + c[2] + c[3] + c[4] + c[5] + c[6] + c[7];
        s += __shfl_xor(s, 16, 32);       // full column sum, same in every lane
        tot[q] = s;
    }

    if (lane == 0) {
        const float invN = 1.0f / (float)N_IMG;
        const float coord = tot[0], objl = tot[1], noobjl = tot[2], classl = tot[3];
        out[0] = (LAMBDA_COORD * coord + objl + LAMBDA_NOOBJ * noobjl + classl) * invN;
        out[1] = coord  * invN;
        out[2] = objl   * invN;
        out[3] = noobjl * invN;
        out[4] = classl * invN;
    }
}

// ---------------- Launch ----------------
extern "C" void kernel_launch(void* const* d_in, const int* in_sizes, int n_in,
                              void* d_out, int out_size, void* d_ws, size_t ws_size,
                              hipStream_t stream)
{
    (void)in_sizes; (void)n_in; (void)out_size; (void)ws_size;
    const float* pred = (const float*)d_in[0];
    const float* tgt  = (const float*)d_in[1];
    float* out = (float*)d_out;
    float* ws  = (float*)d_ws;   // uses NBLOCKS*4 floats = 16 KB, fully overwritten

    yolo_loss_main <<<NBLOCKS, NTHREADS, 0, stream>>>(pred, tgt, ws);
    yolo_loss_final<<<1, 32, 0, stream>>>(ws, out);
}